// MyGAT_35485019799702
// MI455X (gfx1250) — compile-verified
//
#include <hip/hip_runtime.h>
#include <math.h>

#define N_NODES 50000
#define N_EDGES 800000
#define HEADS 4
#define DIM 32
#define HID 128
#define NEG_SLOPE 0.2f

typedef __attribute__((ext_vector_type(2))) float v2f;
typedef __attribute__((ext_vector_type(8))) float v8f;

// ---------- order-preserving float<->uint encoding for atomic max ----------
__device__ __forceinline__ unsigned encf(float x) {
  unsigned u = __float_as_uint(x);
  return (u & 0x80000000u) ? ~u : (u | 0x80000000u);
}
__device__ __forceinline__ float decf(unsigned u) {
  return (u & 0x80000000u) ? __uint_as_float(u & 0x7FFFFFFFu)
                           : __uint_as_float(~u);
}

// ---------- gather node embeddings: h[n,c] = emb[in_feat[n], c] ----------
__global__ __launch_bounds__(256)
void gat_gather_emb(const int* __restrict__ feat, const float* __restrict__ emb,
                    float* __restrict__ h, int total) {
  int i = blockIdx.x * blockDim.x + threadIdx.x;
  if (i >= total) return;
  int n = i >> 7;
  int c = i & 127;
  h[i] = emb[(size_t)feat[n] * 128 + c];
}

// ---------- WMMA fp32 GEMM: out[M,128] = A[M,128] @ W[128,128] + bias ------
// One wave computes a 16x128 strip with V_WMMA_F32_16X16X4_F32.
__global__ __launch_bounds__(256)
void gat_gemm_wmma(const float* __restrict__ A, const float* __restrict__ W,
                   const float* __restrict__ bias, float* __restrict__ out,
                   int M) {
  const int lane = threadIdx.x & 31;
  const int wave = threadIdx.x >> 5;
  const int tile = blockIdx.x * 8 + wave;
  if (tile * 16 >= M) return;           // wave-uniform: EXEC stays all-ones
  const int row0  = tile * 16;
  const int mrow  = lane & 15;          // M (and N) index within tile
  const int khalf = (lane >> 4) << 1;   // lanes 16-31 hold K+2,K+3

  v8f acc[8] = {};                      // 8 column tiles of 16

  const float* arow = A + (size_t)(row0 + mrow) * 128;
  for (int k = 0; k < 128; k += 4) {
    float2 av = *(const float2*)(arow + k + khalf);   // A[m][k+khalf .. +1]
    v2f a; a.x = av.x; a.y = av.y;
    const float* w0 = W + (size_t)(k + khalf) * 128 + mrow;
#pragma unroll
    for (int t = 0; t < 8; ++t) {
      v2f b;
      b.x = w0[t * 16];            // W[k+khalf  ][t*16+mrow]
      b.y = w0[t * 16 + 128];      // W[k+khalf+1][t*16+mrow]
      acc[t] = __builtin_amdgcn_wmma_f32_16x16x4_f32(
          false, a, false, b, (short)0, acc[t], false, false);
    }
  }

  // C/D layout: VGPR r -> M = r + (lane>=16 ? 8 : 0), N = t*16 + (lane&15)
  const int rbase = row0 + ((lane >> 4) << 3);
#pragma unroll
  for (int t = 0; t < 8; ++t) {
    const float bv = bias[t * 16 + mrow];
#pragma unroll
    for (int r = 0; r < 8; ++r)
      out[(size_t)(rbase + r) * 128 + t * 16 + mrow] = acc[t][r] + bv;
  }
}

// ---------- zero attention state ----------
__global__ __launch_bounds__(256)
void gat_init_state(unsigned* __restrict__ maxenc, float* __restrict__ denom,
                    float* __restrict__ accum, int n4, int n128) {
  int i = blockIdx.x * blockDim.x + threadIdx.x;
  if (i < n128) accum[i] = 0.0f;
  if (i < n4) { maxenc[i] = 0u; denom[i] = 0.0f; }  // enc=0 <=> -inf-ish
}

// ---------- edge pass 1: logits + segment max ----------
__global__ __launch_bounds__(256)
void gat_edge_logits(const int* __restrict__ src, const int* __restrict__ dst,
                     const float* __restrict__ fs, const float* __restrict__ fd,
                     const float* __restrict__ attn, float* __restrict__ logits,
                     unsigned* __restrict__ maxenc, int EH) {
  int i = blockIdx.x * blockDim.x + threadIdx.x;
  if (i >= EH) return;
  int e = i >> 2, hd = i & 3;
  int s = src[e], d = dst[e];
  const float* ps = fs + (size_t)s * 128 + hd * 32;
  const float* pd = fd + (size_t)d * 128 + hd * 32;
  const float* pa = attn + hd * 32;
  float acc = 0.0f;
#pragma unroll
  for (int j = 0; j < 32; j += 4) {
    float4 vs = *(const float4*)(ps + j);
    float4 vd = *(const float4*)(pd + j);
    float4 va = *(const float4*)(pa + j);
    float x0 = vs.x + vd.x; x0 = x0 > 0.0f ? x0 : NEG_SLOPE * x0;
    float x1 = vs.y + vd.y; x1 = x1 > 0.0f ? x1 : NEG_SLOPE * x1;
    float x2 = vs.z + vd.z; x2 = x2 > 0.0f ? x2 : NEG_SLOPE * x2;
    float x3 = vs.w + vd.w; x3 = x3 > 0.0f ? x3 : NEG_SLOPE * x3;
    acc += x0 * va.x + x1 * va.y + x2 * va.z + x3 * va.w;
  }
  logits[i] = acc;
  atomicMax(&maxenc[(size_t)d * 4 + hd], encf(acc));
}

// ---------- edge pass 2: exp + segment sum ----------
__global__ __launch_bounds__(256)
void gat_edge_exp(const int* __restrict__ dst, const float* __restrict__ logits,
                  const unsigned* __restrict__ maxenc, float* __restrict__ exb,
                  float* __restrict__ denom, int EH) {
  int i = blockIdx.x * blockDim.x + threadIdx.x;
  if (i >= EH) return;
  int e = i >> 2, hd = i & 3;
  int d = dst[e];
  float m  = decf(maxenc[(size_t)d * 4 + hd]);
  float ex = expf(logits[i] - m);
  exb[i] = ex;
  unsafeAtomicAdd(&denom[(size_t)d * 4 + hd], ex);
}

// ---------- edge pass 3: alpha-weighted scatter of fs[src] ----------
__global__ __launch_bounds__(256)
void gat_edge_scatter(const int* __restrict__ src, const int* __restrict__ dst,
                      const float* __restrict__ fs, const float* __restrict__ exb,
                      const float* __restrict__ denom, float* __restrict__ accum,
                      int EH) {
  int i = blockIdx.x * blockDim.x + threadIdx.x;
  if (i >= EH) return;
  int e = i >> 2, hd = i & 3;
  int s = src[e], d = dst[e];
  float alpha = exb[i] / denom[(size_t)d * 4 + hd];
  const float* ps = fs + (size_t)s * 128 + hd * 32;
  float* pd = accum + (size_t)d * 128 + hd * 32;
#pragma unroll
  for (int j = 0; j < 32; ++j)
    unsafeAtomicAdd(pd + j, ps[j] * alpha);
}

// ---------- ELU ----------
__global__ __launch_bounds__(256)
void gat_elu(const float* __restrict__ x, float* __restrict__ y, int n) {
  int i = blockIdx.x * blockDim.x + threadIdx.x;
  if (i >= n) return;
  float v = x[i];
  y[i] = v > 0.0f ? v : (expf(v) - 1.0f);
}

extern "C" void kernel_launch(void* const* d_in, const int* in_sizes, int n_in,
                              void* d_out, int out_size, void* d_ws, size_t ws_size,
                              hipStream_t stream) {
  (void)in_sizes; (void)n_in; (void)out_size; (void)ws_size;

  const int*   in_feat = (const int*)d_in[0];
  const int*   src = (const int*)d_in[1];
  const int*   dst = (const int*)d_in[2];
  const float* emb = (const float*)d_in[3];
  const float* W1l = (const float*)d_in[4];
  const float* b1l = (const float*)d_in[5];
  const float* W1r = (const float*)d_in[6];
  const float* b1r = (const float*)d_in[7];
  const float* a1  = (const float*)d_in[8];
  const float* W2l = (const float*)d_in[9];
  const float* b2l = (const float*)d_in[10];
  const float* W2r = (const float*)d_in[11];
  const float* b2r = (const float*)d_in[12];
  const float* a2  = (const float*)d_in[13];

  char* p = (char*)d_ws;
  auto take = [&](size_t bytes) -> char* {
    char* r = p; p += (bytes + 255) & ~(size_t)255; return r;
  };
  float*    h      = (float*)take((size_t)N_NODES * HID * 4);
  float*    fs     = (float*)take((size_t)N_NODES * HID * 4);
  float*    fd     = (float*)take((size_t)N_NODES * HID * 4);
  float*    accum  = (float*)take((size_t)N_NODES * HID * 4);
  float*    logits = (float*)take((size_t)N_EDGES * HEADS * 4);
  float*    exb    = (float*)take((size_t)N_EDGES * HEADS * 4);
  unsigned* maxenc = (unsigned*)take((size_t)N_NODES * HEADS * 4);
  float*    denom  = (float*)take((size_t)N_NODES * HEADS * 4);

  const int NH  = N_NODES * HID;    // 6.4M elements
  const int EH  = N_EDGES * HEADS;  // 3.2M (edge,head) pairs
  const int NHd = N_NODES * HEADS;

  dim3 blk(256);
  const int gridNH   = (NH + 255) / 256;
  const int gridEH   = (EH + 255) / 256;
  const int gridGemm = (N_NODES / 16 + 7) / 8;   // 3125 tiles, 8 waves/block

  gat_gather_emb<<<gridNH, blk, 0, stream>>>(in_feat, emb, h, NH);

  // ---- layer 1 ----
  gat_gemm_wmma<<<gridGemm, blk, 0, stream>>>(h, W1l, b1l, fs, N_NODES);
  gat_gemm_wmma<<<gridGemm, blk, 0, stream>>>(h, W1r, b1r, fd, N_NODES);
  gat_init_state<<<gridNH, blk, 0, stream>>>(maxenc, denom, accum, NHd, NH);
  gat_edge_logits<<<gridEH, blk, 0, stream>>>(src, dst, fs, fd, a1, logits, maxenc, EH);
  gat_edge_exp<<<gridEH, blk, 0, stream>>>(dst, logits, maxenc, exb, denom, EH);
  gat_edge_scatter<<<gridEH, blk, 0, stream>>>(src, dst, fs, exb, denom, accum, EH);
  gat_elu<<<gridNH, blk, 0, stream>>>(accum, h, NH);

  // ---- layer 2 ----
  gat_gemm_wmma<<<gridGemm, blk, 0, stream>>>(h, W2l, b2l, fs, N_NODES);
  gat_gemm_wmma<<<gridGemm, blk, 0, stream>>>(h, W2r, b2r, fd, N_NODES);
  gat_init_state<<<gridNH, blk, 0, stream>>>(maxenc, denom, accum, NHd, NH);
  gat_edge_logits<<<gridEH, blk, 0, stream>>>(src, dst, fs, fd, a2, logits, maxenc, EH);
  gat_edge_exp<<<gridEH, blk, 0, stream>>>(dst, logits, maxenc, exb, denom, EH);
  gat_edge_scatter<<<gridEH, blk, 0, stream>>>(src, dst, fs, exb, denom, accum, EH);
  gat_elu<<<gridNH, blk, 0, stream>>>(accum, (float*)d_out, NH);
}